// LossFunc_8383776162038
// MI455X (gfx1250) — compile-verified
//
#include <hip/hip_runtime.h>

// CDNA5 / gfx1250 — wave32. Loss = -(mask * weight * log(gathered)).sum() / B
//
// Strategy (latency-bound gather + reduce):
//   Kernel 1: one thread per (b,t) element (12032 total = 47 blocks x 256).
//             Each thread does ONE scattered 4B gather from the 1.5GB score
//             tensor, computes w*log(s)*mask, then the wave reduces its 32
//             partials with a single V_WMMA_F32_16X16X4_F32 ones-matmul
//             (A = partials in 16x4 layout, B = ones  ->  D[m][n] =
//             partial(m) + partial(m+16); summing the 8 acc VGPRs + one
//             shfl_xor(16) yields the full 32-lane sum). Block partial -> d_ws.
//   Kernel 2: single small block reduces the 47 partials, scales by -1/B,
//             writes the scalar to d_out. No atomics, no pre-zeroing needed.

typedef __attribute__((ext_vector_type(2))) float v2f;
typedef __attribute__((ext_vector_type(8))) float v8f;

// Full 32-lane sum via one WMMA. EXEC must be all ones (guaranteed: called
// unconditionally by every thread of full 256-thread blocks).
__device__ __forceinline__ float wave_sum32_wmma(float x) {
  // A-matrix 16x4 f32 layout (wave32): lane L, VGPR v -> A[M=L%16][K=2*(L/16)+v]
  v2f a;
  a[0] = x;     // K = 0 (lanes 0-15) / K = 2 (lanes 16-31)
  a[1] = 0.0f;  // K = 1 / K = 3 contribute nothing
  // B-matrix 4x16 all ones.
  v2f b;
  b[0] = 1.0f;
  b[1] = 1.0f;
  v8f c = {};
  // 8 args: (neg_a, A, neg_b, B, c_mod, C, reuse_a, reuse_b)
  c = __builtin_amdgcn_wmma_f32_16x16x4_f32(
      /*neg_a=*/false, a, /*neg_b=*/false, b,
      /*c_mod=*/(short)0, c, /*reuse_a=*/false, /*reuse_b=*/false);
  // D[m][n] = partial(m) + partial(m+16), column-invariant.
  // Lane n (0-15) holds rows 0..7 in c[0..7]; lane n+16 holds rows 8..15.
  float s = ((c[0] + c[1]) + (c[2] + c[3])) + ((c[4] + c[5]) + (c[6] + c[7]));
  // low half-sum lives in lanes 0-15, high half-sum in lanes 16-31: combine.
  s += __shfl_xor(s, 16);
  return s;  // total of all 32 lane inputs, present in every lane
}

__global__ void loss_partial_kernel(const float* __restrict__ scores,
                                    const long long* __restrict__ gt,
                                    const long long* __restrict__ lengths,
                                    float* __restrict__ partials,
                                    int n, int T, long long V, float beta) {
  const int i = blockIdx.x * blockDim.x + threadIdx.x;

  float val = 0.0f;
  if (i < n) {
    const int b = i / T;
    const int t = i - b * T;
    const long long g = gt[i];           // ground-truth class, int64
    const long long len = lengths[b];    // valid length, int64
    // the single scattered gather: scores[b, t, g]
    const float sv = scores[(long long)i * V + g];
    const float w = (g == 0) ? 1.0f : beta;
    const float lg = logf(sv);           // scores in (1e-4, 1): safe
    val = (t < (int)len) ? (w * lg) : 0.0f;
  }

  // Uniform control flow from here on: every thread executes the WMMA.
  const float wsum = wave_sum32_wmma(val);

  __shared__ float lds[32];
  const int lane = threadIdx.x & 31;
  const int wave = threadIdx.x >> 5;
  if (lane == 0) lds[wave] = wsum;
  __syncthreads();

  if (threadIdx.x == 0) {
    const int nwaves = (blockDim.x + 31) >> 5;
    float s = 0.0f;
    for (int w = 0; w < nwaves; ++w) s += lds[w];
    partials[blockIdx.x] = s;
  }
}

__global__ void loss_final_kernel(const float* __restrict__ partials, int np,
                                  float* __restrict__ out, float neg_inv_b) {
  float s = 0.0f;
  for (int i = threadIdx.x; i < np; i += blockDim.x) s += partials[i];

  // wave32 shuffle tree
  for (int off = 16; off > 0; off >>= 1) s += __shfl_down(s, off);

  __shared__ float lds[32];
  const int lane = threadIdx.x & 31;
  const int wave = threadIdx.x >> 5;
  if (lane == 0) lds[wave] = s;
  __syncthreads();

  if (threadIdx.x == 0) {
    const int nwaves = (blockDim.x + 31) >> 5;
    float tot = 0.0f;
    for (int w = 0; w < nwaves; ++w) tot += lds[w];
    out[0] = tot * neg_inv_b;  // loss = -sum / B
  }
}

extern "C" void kernel_launch(void* const* d_in, const int* in_sizes, int n_in,
                              void* d_out, int out_size, void* d_ws, size_t ws_size,
                              hipStream_t stream) {
  const float* scores = (const float*)d_in[0];        // [B,T,V] fp32
  const long long* gt = (const long long*)d_in[1];    // [B,T]   int64
  const long long* lengths = (const long long*)d_in[2]; // [B]   int64

  const int BT = in_sizes[1];                 // B*T = 12032
  const int B = in_sizes[2];                  // 64
  const int T = BT / B;                       // 188
  const long long V = (long long)(in_sizes[0] / in_sizes[1]);  // 32000

  float* partials = (float*)d_ws;

  const int threads = 256;                    // 8 wave32 waves
  const int blocks = (BT + threads - 1) / threads;  // 47 (exact fit)

  loss_partial_kernel<<<blocks, threads, 0, stream>>>(
      scores, gt, lengths, partials, BT, T, V, /*beta=*/2.0f);

  loss_final_kernel<<<1, 256, 0, stream>>>(
      partials, blocks, (float*)d_out, -1.0f / (float)B);
}